// Network_13795434955026
// MI455X (gfx1250) — compile-verified
//
#include <hip/hip_runtime.h>
#include <hip/hip_bf16.h>

typedef __attribute__((ext_vector_type(2))) float v2f;
typedef __attribute__((ext_vector_type(4))) float v4f;
typedef __attribute__((ext_vector_type(8))) float v8f;

#define B_   64
#define T_   512
#define D_   768
#define C_   31
#define NSEG 32     // C_ + 1 segments per batch

// ---------------------------------------------------------------------------
// Kernel 1: scores[b,t] = dot(hidden[b,t,:], w5) + b5   (one wave per token)
// ---------------------------------------------------------------------------
__global__ void k_scores(const float* __restrict__ h,
                         const float* __restrict__ w5,
                         const float* __restrict__ b5,
                         float* __restrict__ scores) {
    const int wave = (blockIdx.x * blockDim.x + threadIdx.x) >> 5;  // token id, exact grid
    const int lane = threadIdx.x & 31;
    const float* row = h + (size_t)wave * D_;
    float s = 0.0f;
#pragma unroll
    for (int i = 0; i < D_ / (32 * 4); ++i) {          // 6 float4 per lane, coalesced
        const int off = (i * 32 + lane) * 4;
        v4f hv = *(const v4f*)(row + off);
        v4f wv = *(const v4f*)(w5 + off);
        s += hv.x * wv.x + hv.y * wv.y + hv.z * wv.z + hv.w * wv.w;
    }
#pragma unroll
    for (int m = 16; m >= 1; m >>= 1) s += __shfl_xor(s, m, 32);
    if (lane == 0) scores[wave] = s + b5[0];
}

// ---------------------------------------------------------------------------
// Kernel 2: per-segment softmax -> dense weight matrix W[b, 32, 512]
// One block per batch (32 waves); wave j owns segment j.
// ---------------------------------------------------------------------------
__global__ void k_weights(const float* __restrict__ scores,
                          const int* __restrict__ clause_b,
                          float* __restrict__ W) {
    const int b    = blockIdx.x;
    const int j    = threadIdx.x >> 5;   // segment 0..31
    const int lane = threadIdx.x & 31;

    const int start = (j == 0)        ? 0  : clause_b[b * C_ + j - 1];
    const int end   = (j == NSEG - 1) ? T_ : clause_b[b * C_ + j];

    const float* s = scores + (size_t)b * T_;

    float mx = -1e30f;
    for (int t = start + lane; t < end; t += 32) mx = fmaxf(mx, s[t]);
#pragma unroll
    for (int m = 16; m >= 1; m >>= 1) mx = fmaxf(mx, __shfl_xor(mx, m, 32));

    float sum = 0.0f;
    for (int t = start + lane; t < end; t += 32) sum += expf(s[t] - mx);
#pragma unroll
    for (int m = 16; m >= 1; m >>= 1) sum += __shfl_xor(sum, m, 32);
    const float inv = 1.0f / sum;

    float* wrow = W + ((size_t)b * NSEG + j) * T_;
    for (int t = lane; t < T_; t += 32) {
        float v = 0.0f;
        if (t >= start && t < end) v = expf(s[t] - mx) * inv;
        wrow[t] = v;
    }
}

// ---------------------------------------------------------------------------
// Kernel 3: out[b] = W[b] (32x512) x H[b] (512x768) via V_WMMA_F32_16X16X4_F32
// One wave per (batch, 16-wide N-tile); each wave computes a 32x16 strip
// (both M-tiles) so B-fragments of H are reused across the two WMMAs.
//
// A 16x4 f32 layout: lanes 0-15 -> {K=k0, k0+1}, lanes 16-31 -> {K=k0+2, k0+3}
// B 4x16 f32 layout (mirror): lanes 0-15 -> K=k0/k0+1 at N=lane,
//                             lanes 16-31 -> K=k0+2/k0+3 at N=lane-16
// C/D: VGPR r -> M=r (lanes 0-15), M=8+r (lanes 16-31), N = lane%16
// ---------------------------------------------------------------------------
__global__ void k_pool_wmma(const float* __restrict__ h,
                            const float* __restrict__ W,
                            float* __restrict__ out) {
    const int NT    = D_ / 16;                                        // 48 N-tiles
    const int gwave = (blockIdx.x * blockDim.x + threadIdx.x) >> 5;   // exact grid: 64*48
    const int lane  = threadIdx.x & 31;
    const int b     = gwave / NT;
    const int nt    = gwave % NT;

    const int n0   = nt * 16;
    const int half = lane >> 4;     // 0: lanes 0-15, 1: lanes 16-31
    const int lm   = lane & 15;

    const float* hb = h + (size_t)b * T_ * D_;
    const float* Wb = W + (size_t)b * NSEG * T_;

    v8f acc0 = {};
    v8f acc1 = {};
    const int ncol = n0 + lm;

    for (int k0 = 0; k0 < T_; k0 += 4) {
        const int kk = k0 + 2 * half;
        // A fragments (contiguous float2 from dense weight rows)
        v2f a0 = *(const v2f*)(Wb + (size_t)lm * T_ + kk);            // M-tile 0: rows 0-15
        v2f a1 = *(const v2f*)(Wb + (size_t)(16 + lm) * T_ + kk);     // M-tile 1: rows 16-31
        // B fragment (two rows of H at column ncol)
        v2f bv;
        bv.x = hb[(size_t)kk * D_ + ncol];
        bv.y = hb[(size_t)(kk + 1) * D_ + ncol];

        acc0 = __builtin_amdgcn_wmma_f32_16x16x4_f32(false, a0, false, bv,
                                                     (short)0, acc0, false, false);
        acc1 = __builtin_amdgcn_wmma_f32_16x16x4_f32(false, a1, false, bv,
                                                     (short)0, acc1, false, false);
    }

    float* ob = out + (size_t)b * NSEG * D_;
#pragma unroll
    for (int r = 0; r < 8; ++r) {
        const int m = half * 8 + r;
        ob[(size_t)m * D_ + ncol]        = acc0[r];   // M-tile 0: rows 0-15
        ob[(size_t)(16 + m) * D_ + ncol] = acc1[r];   // M-tile 1: rows 16-31
    }
}

// ---------------------------------------------------------------------------
extern "C" void kernel_launch(void* const* d_in, const int* in_sizes, int n_in,
                              void* d_out, int out_size, void* d_ws, size_t ws_size,
                              hipStream_t stream) {
    const float* h        = (const float*)d_in[0];   // [B, T, D] fp32
    const int*   clause_b = (const int*)d_in[1];     // [B, C]   int32
    const float* w5       = (const float*)d_in[2];   // [D, 1]   fp32
    const float* b5       = (const float*)d_in[3];   // [1]      fp32
    float*       out      = (float*)d_out;           // [B, 32, D] fp32

    float* scores = (float*)d_ws;                                         // B*T floats (128 KB)
    float* Wmat   = (float*)((char*)d_ws + (size_t)B_ * T_ * sizeof(float)); // B*32*512 floats (4 MB)

    // 1) fc5 scores: one wave per token -> 64*512 waves, 8 waves/block
    k_scores<<<(B_ * T_) / 8, 256, 0, stream>>>(h, w5, b5, scores);

    // 2) per-segment softmax -> dense weights: one block per batch, 32 waves
    k_weights<<<B_, NSEG * 32, 0, stream>>>(scores, clause_b, Wmat);

    // 3) batched 32x768x512 GEMM via f32 WMMA: 64*48 waves, 8 waves/block
    k_pool_wmma<<<(B_ * (D_ / 16)) / 8, 256, 0, stream>>>(h, Wmat, out);
}